// GCN_62259845923389
// MI455X (gfx1250) — compile-verified
//
#include <hip/hip_runtime.h>
#include <hip/hip_bf16.h>
#include <math.h>

// ---------------------------------------------------------------------------
// 2-layer GCN for MI455X (gfx1250). GEMMs via V_WMMA_F32_16X16X4_F32 (full
// f32 precision; GEMM is ~2.1 GFLOP total and NOT the bottleneck). Edge
// aggregation is L2-resident scatter-add (H1=25.6MB, H2=16MB << 192MB L2),
// vectorized as float4 loads + 4 atomics per thread to cut redundant
// edge-metadata traffic and use b128 memory ops.
// ---------------------------------------------------------------------------

typedef float v2f __attribute__((ext_vector_type(2)));
typedef float v8f __attribute__((ext_vector_type(8)));

#define N_NODES   100000
#define N_FEAT    128
#define HIDDEN    64
#define N_CLASSES 40
#define N_EDGES   1600000

// ---------------- utility ----------------
__global__ void zero_f32v4(float4* __restrict__ p, long n4) {
  long i = (long)blockIdx.x * blockDim.x + threadIdx.x;
  long stride = (long)gridDim.x * blockDim.x;
  const float4 z = make_float4(0.f, 0.f, 0.f, 0.f);
  for (; i < n4; i += stride) p[i] = z;
}

// ---------------- degree / norm ----------------
__global__ void deg_init(float* __restrict__ deg) {
  int i = blockIdx.x * blockDim.x + threadIdx.x;
  if (i < N_NODES) deg[i] = 1.0f;  // self-loop contributes 1 to every node
}

__global__ void deg_accum(const long long* __restrict__ dst, float* __restrict__ deg) {
  int e = blockIdx.x * blockDim.x + threadIdx.x;
  if (e < N_EDGES) atomicAdd(&deg[(int)dst[e]], 1.0f);
}

__global__ void deg_to_dinv(float* __restrict__ deg) {
  int i = blockIdx.x * blockDim.x + threadIdx.x;
  if (i < N_NODES) deg[i] = rsqrtf(deg[i]);  // deg >= 1 always (self-loops)
}

// ---------------- layer-1 GEMM: H1[100000x64] = X[100000x128] @ W1[128x64] ----
// One wave per 16x16 tile; f32 WMMA, K stepped by 4 (32 v_wmma per tile).
__global__ __launch_bounds__(32) void gemm1_wmma(const float* __restrict__ X,
                                                 const float* __restrict__ W,
                                                 float* __restrict__ H) {
  const int m0   = blockIdx.x * 16;       // 6250 tiles, exact
  const int n0   = blockIdx.y * 16;       // 4 tiles, exact
  const int lane = threadIdx.x & 31;
  const int half = lane >> 4;             // K-group select (A/B), M-group (C/D)
  const int r    = lane & 15;

  const float* arow = X + (size_t)(m0 + r) * N_FEAT;
  v8f acc = {};
  for (int k0 = 0; k0 < N_FEAT; k0 += 4) {
    const int ka = k0 + 2 * half;
    v2f a, b;
    a.x = arow[ka];
    a.y = arow[ka + 1];
    b.x = W[(size_t)ka       * HIDDEN + n0 + r];
    b.y = W[(size_t)(ka + 1) * HIDDEN + n0 + r];
    acc = __builtin_amdgcn_wmma_f32_16x16x4_f32(false, a, false, b,
                                                (short)0, acc, false, false);
  }
  float* outp = H + (size_t)(m0 + 8 * half) * HIDDEN + n0 + r;
#pragma unroll
  for (int v = 0; v < 8; ++v) outp[(size_t)v * HIDDEN] = acc[v];
}

// ---------------- layer-2 GEMM: H2[100000x40] = A[100000x64] @ W2[64x40] -----
__global__ __launch_bounds__(32) void gemm2_wmma(const float* __restrict__ A,
                                                 const float* __restrict__ W,
                                                 float* __restrict__ H) {
  const int m0   = blockIdx.x * 16;
  const int n0   = blockIdx.y * 16;       // 3 tiles: cols 0..47, guard >= 40
  const int lane = threadIdx.x & 31;
  const int half = lane >> 4;
  const int r    = lane & 15;
  const bool colok = (n0 + r) < N_CLASSES;

  const float* arow = A + (size_t)(m0 + r) * HIDDEN;
  v8f acc = {};
  for (int k0 = 0; k0 < HIDDEN; k0 += 4) {
    const int ka = k0 + 2 * half;
    v2f a, b;
    a.x = arow[ka];
    a.y = arow[ka + 1];
    b.x = colok ? W[(size_t)ka       * N_CLASSES + n0 + r] : 0.0f;
    b.y = colok ? W[(size_t)(ka + 1) * N_CLASSES + n0 + r] : 0.0f;
    acc = __builtin_amdgcn_wmma_f32_16x16x4_f32(false, a, false, b,
                                                (short)0, acc, false, false);
  }
  if (colok) {
    float* outp = H + (size_t)(m0 + 8 * half) * N_CLASSES + n0 + r;
#pragma unroll
    for (int v = 0; v < 8; ++v) outp[(size_t)v * N_CLASSES] = acc[v];
  }
}

// ---------------- edge scatter-add (L2-resident atomics, float4 payload) ----
// Layer 1: 16 threads per edge, each owns 4 consecutive feats (b128 load).
__global__ void scatter1(const long long* __restrict__ src,
                         const long long* __restrict__ dst,
                         const float* __restrict__ dinv,
                         const float* __restrict__ H1,
                         float* __restrict__ AGG1) {
  long idx = (long)blockIdx.x * blockDim.x + threadIdx.x;
  if (idx >= (long)N_EDGES * (HIDDEN / 4)) return;
  const int e = (int)(idx >> 4);          // HIDDEN/4 == 16
  const int g = (int)(idx & 15);
  const int s = (int)src[e];
  const int d = (int)dst[e];
  const float w = dinv[s] * dinv[d];
  const float4 h = *(const float4*)(H1 + (size_t)s * HIDDEN + g * 4);
  float* o = AGG1 + (size_t)d * HIDDEN + g * 4;
  atomicAdd(o + 0, h.x * w);
  atomicAdd(o + 1, h.y * w);
  atomicAdd(o + 2, h.z * w);
  atomicAdd(o + 3, h.w * w);
}

// Layer 2: 10 threads per edge (40 feats / 4).
__global__ void scatter2(const long long* __restrict__ src,
                         const long long* __restrict__ dst,
                         const float* __restrict__ dinv,
                         const float* __restrict__ H2,
                         float* __restrict__ OUT) {
  long idx = (long)blockIdx.x * blockDim.x + threadIdx.x;
  if (idx >= (long)N_EDGES * (N_CLASSES / 4)) return;
  const int e = (int)(idx / (N_CLASSES / 4));
  const int g = (int)(idx - (long)e * (N_CLASSES / 4));
  const int s = (int)src[e];
  const int d = (int)dst[e];
  const float w = dinv[s] * dinv[d];
  const float4 h = *(const float4*)(H2 + (size_t)s * N_CLASSES + g * 4);
  float* o = OUT + (size_t)d * N_CLASSES + g * 4;
  atomicAdd(o + 0, h.x * w);
  atomicAdd(o + 1, h.y * w);
  atomicAdd(o + 2, h.z * w);
  atomicAdd(o + 3, h.w * w);
}

// ---------------- fused self-loop + bias + ReLU (layer 1, in place) ---------
__global__ void relu_bias(float* __restrict__ AGG1, const float* __restrict__ H1,
                          const float* __restrict__ dinv, const float* __restrict__ b1) {
  long idx = (long)blockIdx.x * blockDim.x + threadIdx.x;   // N_NODES * 16 groups
  if (idx >= (long)N_NODES * (HIDDEN / 4)) return;
  const int i = (int)(idx >> 4);
  const int g = (int)(idx & 15);
  const float w = dinv[i] * dinv[i];                         // self-loop norm
  const size_t o = (size_t)i * HIDDEN + g * 4;
  const float4 a  = *(const float4*)(AGG1 + o);
  const float4 h  = *(const float4*)(H1 + o);
  const float4 bb = *(const float4*)(b1 + g * 4);
  float4 v;
  v.x = fmaxf(a.x + h.x * w + bb.x, 0.0f);
  v.y = fmaxf(a.y + h.y * w + bb.y, 0.0f);
  v.z = fmaxf(a.z + h.z * w + bb.z, 0.0f);
  v.w = fmaxf(a.w + h.w * w + bb.w, 0.0f);
  *(float4*)(AGG1 + o) = v;
}

// ---------------- fused self-loop + bias + log_softmax (layer 2) ------------
__global__ void finalize_kernel(float* __restrict__ OUT, const float* __restrict__ H2,
                                const float* __restrict__ dinv,
                                const float* __restrict__ b2) {
  int i = blockIdx.x * blockDim.x + threadIdx.x;
  if (i >= N_NODES) return;
  const float w = dinv[i] * dinv[i];
  float v[N_CLASSES];
  float m = -INFINITY;
#pragma unroll
  for (int f = 0; f < N_CLASSES; ++f) {
    const size_t o = (size_t)i * N_CLASSES + f;
    const float t = OUT[o] + H2[o] * w + b2[f];
    v[f] = t;
    m = fmaxf(m, t);
  }
  float s = 0.0f;
#pragma unroll
  for (int f = 0; f < N_CLASSES; ++f) s += expf(v[f] - m);
  const float lse = m + logf(s);
#pragma unroll
  for (int f = 0; f < N_CLASSES; ++f) OUT[(size_t)i * N_CLASSES + f] = v[f] - lse;
}

// ---------------------------------------------------------------------------
extern "C" void kernel_launch(void* const* d_in, const int* in_sizes, int n_in,
                              void* d_out, int out_size, void* d_ws, size_t ws_size,
                              hipStream_t stream) {
  const float*     x  = (const float*)d_in[0];
  const long long* ei = (const long long*)d_in[1];   // int64 per reference
  const float*     W1 = (const float*)d_in[2];
  const float*     b1 = (const float*)d_in[3];
  const float*     W2 = (const float*)d_in[4];
  const float*     b2 = (const float*)d_in[5];
  float* out = (float*)d_out;

  const long long* src = ei;             // edge_index[0]
  const long long* dst = ei + N_EDGES;   // edge_index[1]

  // workspace layout (all sizes 256B-aligned): H1 | AGG1 | H2 | dinv
  char* ws = (char*)d_ws;
  float* H1   = (float*)(ws);
  float* AGG1 = (float*)(ws + (size_t)N_NODES * HIDDEN * sizeof(float));
  float* H2   = (float*)(ws + (size_t)N_NODES * HIDDEN * sizeof(float) * 2);
  float* dinv = (float*)(ws + (size_t)N_NODES * HIDDEN * sizeof(float) * 2
                            + (size_t)N_NODES * N_CLASSES * sizeof(float));

  // degree -> dinv (dinv buffer doubles as deg accumulator)
  deg_init   <<<(N_NODES + 255) / 256, 256, 0, stream>>>(dinv);
  deg_accum  <<<(N_EDGES + 255) / 256, 256, 0, stream>>>(dst, dinv);
  deg_to_dinv<<<(N_NODES + 255) / 256, 256, 0, stream>>>(dinv);

  // zero accumulators (graph-capture safe: plain kernels, b128 stores)
  zero_f32v4<<<1024, 256, 0, stream>>>((float4*)AGG1, (long)N_NODES * HIDDEN / 4);
  zero_f32v4<<<1024, 256, 0, stream>>>((float4*)out,  (long)N_NODES * N_CLASSES / 4);

  // layer 1: WMMA GEMM -> edge scatter -> fused self-loop+bias+ReLU
  dim3 g1(N_NODES / 16, HIDDEN / 16);
  gemm1_wmma<<<g1, 32, 0, stream>>>(x, W1, H1);
  {
    long work = (long)N_EDGES * (HIDDEN / 4);
    scatter1<<<(int)((work + 255) / 256), 256, 0, stream>>>(src, dst, dinv, H1, AGG1);
  }
  {
    long work = (long)N_NODES * (HIDDEN / 4);
    relu_bias<<<(int)((work + 255) / 256), 256, 0, stream>>>(AGG1, H1, dinv, b1);
  }

  // layer 2: WMMA GEMM -> edge scatter -> fused self-loop+bias+log_softmax
  dim3 g2(N_NODES / 16, (N_CLASSES + 15) / 16);
  gemm2_wmma<<<g2, 32, 0, stream>>>(AGG1, W2, H2);
  {
    long work = (long)N_EDGES * (N_CLASSES / 4);
    scatter2<<<(int)((work + 255) / 256), 256, 0, stream>>>(src, dst, dinv, H2, out);
  }
  finalize_kernel<<<(N_NODES + 255) / 256, 256, 0, stream>>>(out, H2, dinv, b2);
}